// GNN_73048803770588
// MI455X (gfx1250) — compile-verified
//
#include <hip/hip_runtime.h>
#include <cstddef>

// Problem constants (match reference)
#define Bx 16
#define Nn 256
#define Dd 128
#define Ll 3
#define Ee 32768
#define NODES (Bx * Nn) // 4096

typedef __attribute__((ext_vector_type(2))) float v2f;
typedef __attribute__((ext_vector_type(8))) float v8f;

__device__ __forceinline__ float silu_f(float x) { return x / (1.0f + __expf(-x)); }

// Block reduction for blockDim.x == 128 (4 wave32's)
__device__ __forceinline__ float blk_reduce_sum(float v, float* lds) {
#pragma unroll
  for (int o = 16; o > 0; o >>= 1) v += __shfl_xor(v, o, 32);
  int wid = threadIdx.x >> 5;
  if ((threadIdx.x & 31) == 0) lds[wid] = v;
  __syncthreads();
  float t = lds[0] + lds[1] + lds[2] + lds[3];
  __syncthreads();
  return t;
}

#define WMMA_F32(a, b, c) \
  __builtin_amdgcn_wmma_f32_16x16x4_f32(false, (a), false, (b), (short)0, (c), false, false)

// ---------------------------------------------------------------------------
// Generic WMMA f32 GEMM: one wave per 16x16 C tile. grid=(N/16, M/16, batch).
// C[m,n] = sum_k A[m,k] * (TRANSB ? B[n,k] : B[k,n])  (+bias[n], optional silu)
// A-fragment (16x4 f32): lane<16 -> K={k0,k0+1}; lane>=16 -> K={k0+2,k0+3}.
// C/D layout: row = vgpr + 8*(lane>=16), col = lane&15.
// K must be a multiple of 16 (unrolled 4 WMMAs per iteration).
// ---------------------------------------------------------------------------
template <int TRANSB>
__global__ void gnn_gemm_wmma(const float* __restrict__ A, int lda, size_t sA,
                              const float* __restrict__ Bm, int ldb, size_t sB,
                              float* __restrict__ C, int ldc, size_t sC,
                              int K, const float* __restrict__ bias, int act) {
  A += (size_t)blockIdx.z * sA;
  Bm += (size_t)blockIdx.z * sB;
  C += (size_t)blockIdx.z * sC;
  const int l = threadIdx.x;
  const int half = l >> 4;
  const int l16 = l & 15;
  const int mrow = blockIdx.y * 16 + l16;
  const int ncol = blockIdx.x * 16 + l16;

  const float* Arow = A + (size_t)mrow * lda + 2 * half;   // contiguous K pairs
  const float* Brow = TRANSB ? (Bm + (size_t)ncol * ldb + 2 * half)
                             : (Bm + (size_t)(2 * half) * ldb + ncol);
  v8f acc = {};
  for (int k0 = 0; k0 < K; k0 += 16) {
    v2f a[4], b[4];
#pragma unroll
    for (int u = 0; u < 4; ++u) {
      const int ka = k0 + 4 * u;
      a[u] = *(const v2f*)(Arow + ka);
      if (TRANSB) {
        b[u] = *(const v2f*)(Brow + ka);
      } else {
        b[u].x = Brow[(size_t)ka * ldb];
        b[u].y = Brow[(size_t)(ka + 1) * ldb];
      }
    }
#pragma unroll
    for (int u = 0; u < 4; ++u) acc = WMMA_F32(a[u], b[u], acc);
  }
  const float bn = bias ? bias[ncol] : 0.0f;
#pragma unroll
  for (int v = 0; v < 8; ++v) {
    int m = blockIdx.y * 16 + v + 8 * half;
    float val = acc[v] + bn;
    if (act) val = silu_f(val);
    C[(size_t)m * ldc + ncol] = val;
  }
}

// ---------------------------------------------------------------------------
// Encoder: h[r,d] = x[r]*enc_W[d] + enc_b[d]
// ---------------------------------------------------------------------------
__global__ void gnn_encode(const float* __restrict__ x,
                           const float* __restrict__ encW,
                           const float* __restrict__ encb,
                           float* __restrict__ h) {
  int r = blockIdx.x, d = threadIdx.x;
  h[(size_t)r * Dd + d] = x[r] * encW[d] + encb[d];
}

__global__ void gnn_zero(float* __restrict__ p, int n) {
  int i = blockIdx.x * blockDim.x + threadIdx.x;
  if (i < n) p[i] = 0.0f;
}

// Symmetric adjacency directly: adj + adj^T (diagonal doubled), matching ref.
__global__ void gnn_build_adj(const int* __restrict__ ei,
                              const float* __restrict__ ea,
                              float* __restrict__ adj) {
  int e = blockIdx.x * blockDim.x + threadIdx.x;
  if (e >= Ee) return;
  int i0 = ei[e], i1 = ei[Ee + e];
  int b = i0 / Nn, src = i0 % Nn, dst = i1 % Nn;
  float w = ea[e];
  atomicAdd(&adj[(size_t)b * Nn * Nn + (size_t)src * Nn + dst], w);
  atomicAdd(&adj[(size_t)b * Nn * Nn + (size_t)dst * Nn + src], w);
}

// ---------------------------------------------------------------------------
// Per-node scalar moments for the LayerNorm decomposition.
// stats[r] = {Ti, T2, TW, Sj, S2, SW, 0, 0}
// ---------------------------------------------------------------------------
__global__ void gnn_node_stats(const float* __restrict__ tbuf,
                               const float* __restrict__ sbuf,
                               const float* __restrict__ wj,
                               float* __restrict__ stats) {
  int r = blockIdx.x * blockDim.x + threadIdx.x;
  if (r >= NODES) return;
  const float* t = tbuf + (size_t)r * Dd;
  const float* s = sbuf + (size_t)r * Dd;
  float Ti = 0, T2 = 0, TW = 0, Sj = 0, S2 = 0, SW = 0;
  for (int d = 0; d < Dd; ++d) {
    float td = t[d], sd = s[d], w = wj[d];
    Ti += td; T2 += td * td; TW += td * w;
    Sj += sd; S2 += sd * sd; SW += sd * w;
  }
  stats[r * 8 + 0] = Ti; stats[r * 8 + 1] = T2; stats[r * 8 + 2] = TW;
  stats[r * 8 + 3] = Sj; stats[r * 8 + 4] = S2; stats[r * 8 + 5] = SW;
  stats[r * 8 + 6] = 0.f; stats[r * 8 + 7] = 0.f;
}

// ---------------------------------------------------------------------------
// Fused message + LN + silu + masked aggregation.
// One block (128 threads) per node (b,i); thread owns feature d.
// No barriers inside the j-loop: mean/var come from precomputed scalars + G.
// Writes hcat[r] = [h_row , agg_row] for the update GEMM.
// ---------------------------------------------------------------------------
__global__ void gnn_msg_agg(const float* __restrict__ h,
                            const float* __restrict__ tbuf,
                            const float* __restrict__ sbuf,
                            const float* __restrict__ stats,
                            const float* __restrict__ adj,
                            const float* __restrict__ G,
                            const float* __restrict__ wj,
                            const float* __restrict__ g,
                            const float* __restrict__ beta,
                            float* __restrict__ hcat) {
  __shared__ float lds[4];
  const int r = blockIdx.x;  // node row 0..4095
  const int b = r >> 8;
  const int i = r & 255;
  const int d = threadIdx.x; // 0..127

  const float wd = wj[d];
  const float cd = tbuf[(size_t)r * Dd + d]; // t' = h@Wt + msg_b  (== c_d)
  const float gd = g[d];
  const float bd = beta[d];

  const float Wsum = blk_reduce_sum(wd, lds);
  const float W2 = blk_reduce_sum(wd * wd, lds);

  const float Ti = stats[r * 8 + 0];
  const float T2 = stats[r * 8 + 1];
  const float TW = stats[r * 8 + 2];

  const float* srow = sbuf + (size_t)b * Nn * Dd;
  const float* arow = adj + (size_t)b * Nn * Nn + (size_t)i * Nn;
  const float* grow = G + (size_t)b * Nn * Nn + (size_t)i * Nn;
  const float* st = stats + (size_t)b * Nn * 8;

  float agg = 0.0f;
  for (int j = 0; j < Nn; ++j) {
    const float Sj = st[j * 8 + 3];
    const float S2 = st[j * 8 + 4];
    const float SW = st[j * 8 + 5];
    const float a = arow[j];
    const float Gij = grow[j];
    const float sumv = Ti + Sj + a * Wsum;
    const float mean = sumv * (1.0f / Dd);
    const float sumv2 = T2 + S2 + a * a * W2 + 2.0f * Gij + 2.0f * a * (TW + SW);
    const float var = sumv2 * (1.0f / Dd) - mean * mean;
    const float rstd = rsqrtf(var + 1e-5f);
    const float v = cd + srow[(size_t)j * Dd + d] + a * wd;
    const float t = (v - mean) * rstd * gd + bd;
    const float msg = silu_f(t);
    agg += (j != i) ? msg : 0.0f;
  }
  hcat[(size_t)r * 2 * Dd + d] = h[(size_t)r * Dd + d];
  hcat[(size_t)r * 2 * Dd + Dd + d] = agg;
}

// ---------------------------------------------------------------------------
// h += silu(LN(u))   (block per row, thread per feature)
// ---------------------------------------------------------------------------
__global__ void gnn_update(float* __restrict__ h, const float* __restrict__ ubuf,
                           const float* __restrict__ g,
                           const float* __restrict__ beta) {
  __shared__ float lds[4];
  int r = blockIdx.x, d = threadIdx.x;
  float u = ubuf[(size_t)r * Dd + d];
  float s1 = blk_reduce_sum(u, lds);
  float s2 = blk_reduce_sum(u * u, lds);
  float mean = s1 * (1.0f / Dd);
  float var = s2 * (1.0f / Dd) - mean * mean;
  float rstd = rsqrtf(var + 1e-5f);
  float t = (u - mean) * rstd * g[d] + beta[d];
  h[(size_t)r * Dd + d] += silu_f(t);
}

// ---------------------------------------------------------------------------
// Decoder layer 1 as WMMA with A gathered on the fly:
// A[m,k] = k<128 ? h[src_m,k] : h[dst_m,k-128].
// K=256 split into two branch-free 128-halves (h_src then h_dst);
// the 257th input (edge_attr) folded into the epilogue; +bias, silu.
// ---------------------------------------------------------------------------
__global__ void gnn_dec1_wmma(const float* __restrict__ h,
                              const int* __restrict__ ei,
                              const float* __restrict__ ea,
                              const float* __restrict__ W1, // 257 x 128
                              const float* __restrict__ b1,
                              float* __restrict__ e1) { // E x 128
  const int l = threadIdx.x;
  const int half = l >> 4;
  const int l16 = l & 15;
  const int m = blockIdx.y * 16 + l16; // edge (A row) for this lane
  const int n = blockIdx.x * 16 + l16; // out col for this lane
  const int r0 = ei[m];
  const int r1 = ei[Ee + m];
  const float* h0 = h + (size_t)r0 * Dd + 2 * half;
  const float* h1 = h + (size_t)r1 * Dd + 2 * half;
  const float* Bp0 = W1 + (size_t)(2 * half) * Dd + n;        // rows 0..127
  const float* Bp1 = W1 + (size_t)(Dd + 2 * half) * Dd + n;   // rows 128..255
  v8f acc = {};
  // K half 1: A columns 0..127 come from h[src]
  for (int k0 = 0; k0 < Dd; k0 += 16) {
    v2f a[4], b[4];
#pragma unroll
    for (int u = 0; u < 4; ++u) {
      const int ka = k0 + 4 * u;
      a[u] = *(const v2f*)(h0 + ka);
      b[u].x = Bp0[(size_t)ka * Dd];
      b[u].y = Bp0[(size_t)(ka + 1) * Dd];
    }
#pragma unroll
    for (int u = 0; u < 4; ++u) acc = WMMA_F32(a[u], b[u], acc);
  }
  // K half 2: A columns 128..255 come from h[dst]
  for (int k0 = 0; k0 < Dd; k0 += 16) {
    v2f a[4], b[4];
#pragma unroll
    for (int u = 0; u < 4; ++u) {
      const int ka = k0 + 4 * u;
      a[u] = *(const v2f*)(h1 + ka);
      b[u].x = Bp1[(size_t)ka * Dd];
      b[u].y = Bp1[(size_t)(ka + 1) * Dd];
    }
#pragma unroll
    for (int u = 0; u < 4; ++u) acc = WMMA_F32(a[u], b[u], acc);
  }
  const float wlast = W1[(size_t)(2 * Dd) * Dd + n];
  const float bn = b1[n];
#pragma unroll
  for (int v = 0; v < 8; ++v) {
    int mr = blockIdx.y * 16 + v + 8 * half;
    float val = acc[v] + ea[mr] * wlast + bn;
    e1[(size_t)mr * Dd + n] = silu_f(val);
  }
}

// Decoder layer 3: per-edge 64-dot + tanh
__global__ void gnn_dec3(const float* __restrict__ e2,
                         const float* __restrict__ W3,
                         const float* __restrict__ b3,
                         float* __restrict__ out) {
  int m = blockIdx.x * blockDim.x + threadIdx.x;
  if (m >= Ee) return;
  float acc = b3[0];
  const float* row = e2 + (size_t)m * 64;
#pragma unroll
  for (int k = 0; k < 64; ++k) acc += row[k] * W3[k];
  out[m] = tanhf(acc);
}

// ---------------------------------------------------------------------------
extern "C" void kernel_launch(void* const* d_in, const int* in_sizes, int n_in,
                              void* d_out, int out_size, void* d_ws,
                              size_t ws_size, hipStream_t stream) {
  (void)in_sizes; (void)n_in; (void)out_size; (void)ws_size;
  const float* x = (const float*)d_in[0];
  const int* ei = (const int*)d_in[1];
  const float* ea = (const float*)d_in[2];
  // d_in[3] (batch) derivable from edge_index; unused.
  const float* encW = (const float*)d_in[4];
  const float* encb = (const float*)d_in[5];
  const float* msgW = (const float*)d_in[6];   // (L, 257, 128)
  const float* msgb = (const float*)d_in[7];
  const float* msgg = (const float*)d_in[8];
  const float* msgbt = (const float*)d_in[9];
  const float* updW = (const float*)d_in[10];  // (L, 256, 128)
  const float* updb = (const float*)d_in[11];
  const float* updg = (const float*)d_in[12];
  const float* updbt = (const float*)d_in[13];
  const float* dW1 = (const float*)d_in[14];
  const float* db1 = (const float*)d_in[15];
  const float* dW2 = (const float*)d_in[16];
  const float* db2 = (const float*)d_in[17];
  const float* dW3 = (const float*)d_in[18];
  const float* db3 = (const float*)d_in[19];
  float* out = (float*)d_out;

  float* ws = (float*)d_ws;
  float* h = ws;                         // 4096*128
  float* tbuf = h + NODES * Dd;          // 4096*128
  float* sbuf = tbuf + NODES * Dd;       // 4096*128
  float* hcat = sbuf + NODES * Dd;       // 4096*256
  float* ubuf = hcat + NODES * 2 * Dd;   // 4096*128
  float* adj = ubuf + NODES * Dd;        // 16*256*256
  float* G = adj + Bx * Nn * Nn;         // 16*256*256
  float* stats = G + Bx * Nn * Nn;       // 4096*8
  float* e1 = stats + NODES * 8;         // 32768*128
  float* e2 = e1 + (size_t)Ee * Dd;      // 32768*64

  gnn_encode<<<NODES, Dd, 0, stream>>>(x, encW, encb, h);
  gnn_zero<<<(Bx * Nn * Nn) / 256, 256, 0, stream>>>(adj, Bx * Nn * Nn);
  gnn_build_adj<<<Ee / 256, 256, 0, stream>>>(ei, ea, adj);

  for (int l = 0; l < Ll; ++l) {
    const float* Wl = msgW + (size_t)l * (2 * Dd + 1) * Dd;
    const float* wj = Wl + 2 * Dd * Dd; // row 256 of msg_W[l]
    // t' = h @ Wt + msg_b  (this is c_i)
    gnn_gemm_wmma<0><<<dim3(Dd / 16, NODES / 16, 1), 32, 0, stream>>>(
        h, Dd, 0, Wl, Dd, 0, tbuf, Dd, 0, Dd, msgb + l * Dd, 0);
    // s = h @ Ws
    gnn_gemm_wmma<0><<<dim3(Dd / 16, NODES / 16, 1), 32, 0, stream>>>(
        h, Dd, 0, Wl + Dd * Dd, Dd, 0, sbuf, Dd, 0, Dd,
        (const float*)nullptr, 0);
    gnn_node_stats<<<NODES / 128, 128, 0, stream>>>(tbuf, sbuf, wj, stats);
    // G[b] = T_b @ S_b^T  (batched, B transposed)
    gnn_gemm_wmma<1><<<dim3(Nn / 16, Nn / 16, Bx), 32, 0, stream>>>(
        tbuf, Dd, (size_t)Nn * Dd, sbuf, Dd, (size_t)Nn * Dd, G, Nn,
        (size_t)Nn * Nn, Dd, (const float*)nullptr, 0);
    gnn_msg_agg<<<NODES, Dd, 0, stream>>>(h, tbuf, sbuf, stats, adj, G, wj,
                                          msgg + l * Dd, msgbt + l * Dd, hcat);
    // u = [h ; agg] @ upd_W + upd_b
    gnn_gemm_wmma<0><<<dim3(Dd / 16, NODES / 16, 1), 32, 0, stream>>>(
        hcat, 2 * Dd, 0, updW + (size_t)l * 2 * Dd * Dd, Dd, 0, ubuf, Dd, 0,
        2 * Dd, updb + l * Dd, 0);
    gnn_update<<<NODES, Dd, 0, stream>>>(h, ubuf, updg + l * Dd,
                                         updbt + l * Dd);
  }

  gnn_dec1_wmma<<<dim3(Dd / 16, Ee / 16), 32, 0, stream>>>(h, ei, ea, dW1, db1,
                                                           e1);
  // e2 = silu(e1 @ dec_W2 + dec_b2)
  gnn_gemm_wmma<0><<<dim3(64 / 16, Ee / 16, 1), 32, 0, stream>>>(
      e1, Dd, 0, dW2, 64, 0, e2, 64, 0, Dd, db2, 1);
  gnn_dec3<<<Ee / 256, 256, 0, stream>>>(e2, dW3, db3, out);
}